// mmld_Loss_21010980012214
// MI455X (gfx1250) — compile-verified
//
#include <hip/hip_runtime.h>
#include <math.h>

// ---------------------------------------------------------------------------
// Problem geometry (from reference): B=16, S=2048, P=256, D=3
//   rows   = B*S = 32768 ; rowlen = P*D = 768 floats = 3072 B/tensor/row
// Memory-bound: 4 x 100.7MB read -> ~17.3us floor @ 23.3 TB/s.
// Strategy: one wave per row, coalesced b128 streaming, fp32 throughout,
// WMMA f32 16x16x4 (B=ones) for the exact cross-lane sum, __all() for the
// mask, permlanex16 for the half-wave fold (no LDS round-trips).
// ---------------------------------------------------------------------------
#define NROWS        (16 * 2048)
#define ROWLEN       768
#define WAVES_PER_BLOCK 8
#define BLOCK_THREADS   256
#define MAX_BLOCKS      2048
#define L2_LAMBDA       0.01f

typedef float v2f  __attribute__((ext_vector_type(2)));
typedef float v8f  __attribute__((ext_vector_type(8)));

// Half-wave (xor-16) swap-and-add. v_permlanex16_b32 with identity lane
// selects swaps lanes 0-15 <-> 16-31 entirely in VALU (no LDS, no dscnt).
__device__ __forceinline__ float xor16_add(float s) {
#if __has_builtin(__builtin_amdgcn_permlanex16)
    int si = __float_as_int(s);
    int sw = __builtin_amdgcn_permlanex16(si, si, 0x76543210, 0xFEDCBA98,
                                          false, false);
    return s + __int_as_float(sw);
#else
    return s + __shfl_xor(s, 16, 32);
#endif
}

// ---------------------------------------------------------------------------
// Wave-wide exact fp32 sum via V_WMMA_F32_16X16X4_F32 (confirmed lowering).
//   A = 16x4 f32 over (2 VGPRs x 32 lanes): per-lane partial in slot0, zero
//   in slot1 -> sum over ALL A slots == sum over lanes (layout independent).
//   B = all-ones (uniform -> layout independent). D[m][n] = sum_k A[m][k];
//   per-lane fold of D's 8 VGPRs gives the M=0..7 / M=8..15 half-sums in the
//   two lane halves; one xor-16 fold completes the total in every lane.
// ---------------------------------------------------------------------------
__device__ __forceinline__ float wave_sum_wmma(float p) {
    v2f a; a.x = p;    a.y = 0.0f;
    v2f b; b.x = 1.0f; b.y = 1.0f;
    v8f c = {};
    // 8 args: (neg_a, A, neg_b, B, c_mod, C, reuse_a, reuse_b)
    c = __builtin_amdgcn_wmma_f32_16x16x4_f32(false, a, false, b,
                                              (short)0, c, false, false);
    float s = ((c[0] + c[1]) + (c[2] + c[3])) + ((c[4] + c[5]) + (c[6] + c[7]));
    return xor16_add(s);
}

__device__ __forceinline__ void acc_sq(const float4 x, const float4 t, float& p) {
    float d0 = x.x - t.x, d1 = x.y - t.y, d2 = x.z - t.z, d3 = x.w - t.w;
    p = fmaf(d0, d0, p);
    p = fmaf(d1, d1, p);
    p = fmaf(d2, d2, p);
    p = fmaf(d3, d3, p);
}

// ---------------------------------------------------------------------------
// Kernel 1: one wave per row. Lane l reads float4 at element (j*128 + 4*l),
// j=0..5 -> 768 floats/row/tensor, coalesced 512B bursts per wave.
// Mask: min(target_row) >= 0  <=>  __all(per-lane tmin >= 0)  (no LDS tree).
// Deterministic: fixed row ownership, fixed in-block order, no float atomics.
// ---------------------------------------------------------------------------
__global__ __launch_bounds__(BLOCK_THREADS)
void mmld_loss_rows_kernel(const float* __restrict__ out,
                           const float* __restrict__ gt1,
                           const float* __restrict__ gt2,
                           const float* __restrict__ tgt,
                           float* __restrict__ partials) {
    __shared__ float lds[WAVES_PER_BLOCK];
    const int lane  = threadIdx.x & 31;
    const int wave  = threadIdx.x >> 5;
    const int gwave = blockIdx.x * WAVES_PER_BLOCK + wave;
    const int nwave = gridDim.x * WAVES_PER_BLOCK;

    float acc = 0.0f;
    for (int row = gwave; row < NROWS; row += nwave) {
        const size_t base = (size_t)row * ROWLEN + (size_t)(lane << 2);
        float p0 = 0.0f, p1 = 0.0f, p2 = 0.0f;
        float tmin = 3.402823466e38f;
#pragma unroll
        for (int j = 0; j < 6; ++j) {
            const size_t off = base + (size_t)j * 128;
            const float4 t = *(const float4*)(tgt + off);
            const float4 o = *(const float4*)(out + off);
            const float4 a = *(const float4*)(gt1 + off);
            const float4 b = *(const float4*)(gt2 + off);
            acc_sq(o, t, p0);
            acc_sq(a, t, p1);
            acc_sq(b, t, p2);
            tmin = fminf(tmin, fminf(fminf(t.x, t.y), fminf(t.z, t.w)));
        }
        // Wave-uniform control flow here => EXEC all ones for WMMA.
        const float s0 = wave_sum_wmma(p0);
        const float s1 = wave_sum_wmma(p1);
        const float s2 = wave_sum_wmma(p2);
        const float mflag = __all(tmin >= 0.0f) ? 1.0f : 0.0f;
        acc += mflag * (sqrtf(s0) + 0.1f * sqrtf(s1) + 0.2f * sqrtf(s2));
    }
    if (lane == 0) lds[wave] = acc;
    __syncthreads();
    if (threadIdx.x == 0) {
        float b = 0.0f;
#pragma unroll
        for (int w = 0; w < WAVES_PER_BLOCK; ++w) b += lds[w];
        partials[blockIdx.x] = b;
    }
}

// ---------------------------------------------------------------------------
// Kernel 2: deterministic fixed-tree reduction of block partials + detached
// scalar terms: L2*0.2*l_dynamic*leg + gt_pre0*L2*l_dynamic.
// ---------------------------------------------------------------------------
__global__ __launch_bounds__(BLOCK_THREADS)
void mmld_loss_finalize_kernel(const float* __restrict__ partials, int npart,
                               const float* __restrict__ gt_pre0,
                               const float* __restrict__ leg,
                               const float* __restrict__ l_dyn,
                               float* __restrict__ result) {
    __shared__ float lds[BLOCK_THREADS];
    float s = 0.0f;
    for (int i = threadIdx.x; i < npart; i += BLOCK_THREADS) s += partials[i];
    lds[threadIdx.x] = s;
    __syncthreads();
#pragma unroll
    for (int step = BLOCK_THREADS / 2; step > 0; step >>= 1) {
        if ((int)threadIdx.x < step) lds[threadIdx.x] += lds[threadIdx.x + step];
        __syncthreads();
    }
    if (threadIdx.x == 0) {
        const float ld = l_dyn[0];
        result[0] = lds[0]
                  + L2_LAMBDA * 0.2f * ld * leg[0]
                  + gt_pre0[0] * L2_LAMBDA * ld;
    }
}

extern "C" void kernel_launch(void* const* d_in, const int* in_sizes, int n_in,
                              void* d_out, int out_size, void* d_ws, size_t ws_size,
                              hipStream_t stream) {
    (void)in_sizes; (void)n_in; (void)out_size;
    const float* out_t   = (const float*)d_in[0];
    const float* gt1_t   = (const float*)d_in[1];
    const float* gt2_t   = (const float*)d_in[2];
    const float* tgt_t   = (const float*)d_in[3];
    const float* gt_pre0 = (const float*)d_in[4];
    const float* leg     = (const float*)d_in[5];
    const float* l_dyn   = (const float*)d_in[6];

    float* partials = (float*)d_ws;
    int blocks = MAX_BLOCKS;
    size_t cap = ws_size / sizeof(float);
    if ((size_t)blocks > cap) blocks = (int)(cap ? cap : 1);
    if (blocks < 1) blocks = 1;

    mmld_loss_rows_kernel<<<blocks, BLOCK_THREADS, 0, stream>>>(
        out_t, gt1_t, gt2_t, tgt_t, partials);
    mmld_loss_finalize_kernel<<<1, BLOCK_THREADS, 0, stream>>>(
        partials, blocks, gt_pre0, leg, l_dyn, (float*)d_out);
}